// HeteroGNN_11811160064003
// MI455X (gfx1250) — compile-verified
//
#include <hip/hip_runtime.h>
#include <hip/hip_bf16.h>
#include <math.h>

namespace {

constexpr int NNODES = 1024;
constexpr int NH     = 2;     // attention heads

typedef __attribute__((ext_vector_type(2))) float v2f;
typedef __attribute__((ext_vector_type(8))) float v8f;

__global__ void fill_kernel(float* __restrict__ p, float v, int n) {
  int i = blockIdx.x * blockDim.x + threadIdx.x;
  if (i < n) p[i] = v;
}

// C[M,N] = A[M,K] @ B[K,N]  (fp32, row-major) using V_WMMA_F32_16X16X4_F32.
// One wave -> one 16x16 tile. A frag: lanes 0-15 K={k,k+1}, lanes 16-31 K={k+2,k+3}.
__global__ void gemm_wmma_f32(const float* __restrict__ A, const float* __restrict__ B,
                              float* __restrict__ C, int M, int K, int N) {
  const int wave   = threadIdx.x >> 5;
  const int lane   = threadIdx.x & 31;
  const int tile   = blockIdx.x * (blockDim.x >> 5) + wave;
  const int tilesN = N >> 4;
  if (tile >= (M >> 4) * tilesN) return;      // whole-wave exit -> EXEC all-ones inside
  const int tm  = tile / tilesN;
  const int tn  = tile - tm * tilesN;
  const int row = (tm << 4) + (lane & 15);
  const int col = (tn << 4) + (lane & 15);
  const int khalf = (lane >> 4) << 1;         // 0 for lanes 0-15, 2 for lanes 16-31

  v8f acc = {};
  for (int k = 0; k < K; k += 4) {
    const int kk = k + khalf;
    v2f a = *(const v2f*)(A + (size_t)row * K + kk);   // 8B-aligned global_load_b64
    v2f b;
    b.x = B[(size_t)kk * N + col];
    b.y = B[(size_t)(kk + 1) * N + col];
    acc = __builtin_amdgcn_wmma_f32_16x16x4_f32(false, a, false, b,
                                                (short)0, acc, false, false);
  }
  const int rbase = (tm << 4) + ((lane >> 4) << 3);    // rows 0..7 / 8..15
#pragma unroll
  for (int i = 0; i < 8; ++i)
    C[(size_t)(rbase + i) * N + col] = acc[i];
}

// al_s[n,h] = sum_c xp[n, h*C + c] * a_src[h*C + c]   (and same for al_d)
__global__ void compute_al(const float* __restrict__ xp, const float* __restrict__ a_src,
                           const float* __restrict__ a_dst, float* __restrict__ al_s,
                           float* __restrict__ al_d, int n, int C) {
  int idx = blockIdx.x * blockDim.x + threadIdx.x;
  if (idx >= n * NH) return;
  int node = idx / NH, h = idx - node * NH;
  const float* xr = xp + (size_t)node * NH * C + (size_t)h * C;
  const float* as = a_src + (size_t)h * C;
  const float* ad = a_dst + (size_t)h * C;
  float ss = 0.f, sd = 0.f;
  for (int c = 0; c < C; ++c) { float v = xr[c]; ss += v * as[c]; sd += v * ad[c]; }
  al_s[idx] = ss;
  al_d[idx] = sd;
}

__device__ inline float leaky(float a) { return a > 0.f ? a : 0.2f * a; }

__device__ inline void atomic_max_f32(float* addr, float val) {
  // sign-aware int/uint trick; buffer initialized to -inf (0xff800000)
  if (val >= 0.f) atomicMax((int*)addr, __float_as_int(val));
  else            atomicMin((unsigned int*)addr, __float_as_uint(val));
}

__device__ inline void edge_endpoints(const int* __restrict__ ei, int E, int e,
                                      int& src, int& dst) {
  if (e < E) { src = ei[e]; dst = ei[E + e]; }     // ei[0,:] then ei[1,:]
  else       { src = dst = e - E; }                // appended self-loops
}

__global__ void edge_max(const int* __restrict__ ei, int E, int n,
                         const float* __restrict__ al_s, const float* __restrict__ al_d,
                         float* __restrict__ m) {
  int idx = blockIdx.x * blockDim.x + threadIdx.x;
  if (idx >= (E + n) * NH) return;
  int e = idx / NH, h = idx - e * NH;
  int src, dst; edge_endpoints(ei, E, e, src, dst);
  float a = leaky(al_s[src * NH + h] + al_d[dst * NH + h]);
  atomic_max_f32(&m[dst * NH + h], a);
}

__global__ void edge_sum(const int* __restrict__ ei, int E, int n,
                         const float* __restrict__ al_s, const float* __restrict__ al_d,
                         const float* __restrict__ m, float* __restrict__ coef,
                         float* __restrict__ s) {
  int idx = blockIdx.x * blockDim.x + threadIdx.x;
  if (idx >= (E + n) * NH) return;
  int e = idx / NH, h = idx - e * NH;
  int src, dst; edge_endpoints(ei, E, e, src, dst);
  float a  = leaky(al_s[src * NH + h] + al_d[dst * NH + h]);
  float ev = expf(a - m[dst * NH + h]);
  coef[idx] = ev;
  atomicAdd(&s[dst * NH + h], ev);
}

__global__ void coef_norm(const int* __restrict__ ei, int E, int n,
                          float* __restrict__ coef, const float* __restrict__ s) {
  int idx = blockIdx.x * blockDim.x + threadIdx.x;
  if (idx >= (E + n) * NH) return;
  int e = idx / NH, h = idx - e * NH;
  int src, dst; edge_endpoints(ei, E, e, src, dst);
  (void)src;
  coef[idx] = coef[idx] / (s[dst * NH + h] + 1e-16f);
}

// out[dst, f] += xp[src, f] * coef[e, f/C]  over all edges, F = NH*C features
__global__ void edge_agg(const int* __restrict__ ei, int E, int n, int C,
                         const float* __restrict__ xp, const float* __restrict__ coef,
                         float* __restrict__ out) {
  const int F = NH * C;
  int idx = blockIdx.x * blockDim.x + threadIdx.x;
  if (idx >= (E + n) * F) return;
  int e = idx / F, f = idx - e * F, h = f / C;
  int src, dst; edge_endpoints(ei, E, e, src, dst);
  atomicAdd(&out[(size_t)dst * F + f], xp[(size_t)src * F + f] * coef[e * NH + h]);
}

__global__ void bias_act(float* __restrict__ buf, const float* __restrict__ b0,
                         const float* __restrict__ b1, int total, int F, int do_relu) {
  int i = blockIdx.x * blockDim.x + threadIdx.x;
  if (i >= total) return;
  int f = i % F;
  float v = buf[i] + b0[f] + b1[f];
  buf[i] = do_relu ? fmaxf(v, 0.f) : v;
}

// s_node[n] = h2[n,:] . w_lin   (one wave32 per node)
__global__ void node_dot(const float* __restrict__ h2, const float* __restrict__ w,
                         float* __restrict__ s_node, int n, int F) {
  int wave = threadIdx.x >> 5, lane = threadIdx.x & 31;
  int node = blockIdx.x * (blockDim.x >> 5) + wave;
  if (node >= n) return;
  float sum = 0.f;
  for (int f = lane; f < F; f += 32) sum += h2[(size_t)node * F + f] * w[f];
#pragma unroll
  for (int off = 16; off > 0; off >>= 1) sum += __shfl_xor(sum, off, 32);
  if (lane == 0) s_node[node] = sum;
}

// out[i*n + j] = s[i] + s[j] + b  — factorized pairwise stage, float4 writes
__global__ void pair_out(const float* __restrict__ s_node, const float* __restrict__ b_lin,
                         float* __restrict__ out, int n) {
  int nq = n >> 2;
  int idx = blockIdx.x * blockDim.x + threadIdx.x;
  if (idx >= n * nq) return;
  int i = idx / nq, j4 = (idx - i * nq) << 2;
  float base = s_node[i] + b_lin[0];
  float4 sj = *(const float4*)(s_node + j4);
  float4 o = make_float4(base + sj.x, base + sj.y, base + sj.z, base + sj.w);
  *(float4*)(out + (size_t)i * n + j4) = o;
}

inline int nblk(int total, int per) { return (total + per - 1) / per; }

}  // namespace

extern "C" void kernel_launch(void* const* d_in, const int* in_sizes, int n_in,
                              void* d_out, int out_size, void* d_ws, size_t ws_size,
                              hipStream_t stream) {
  const float* x      = (const float*)d_in[0];
  const int*   ei[2]  = {(const int*)d_in[1], (const int*)d_in[2]};
  const float* W1[2]  = {(const float*)d_in[3],  (const float*)d_in[7]};
  const float* as1[2] = {(const float*)d_in[4],  (const float*)d_in[8]};
  const float* ad1[2] = {(const float*)d_in[5],  (const float*)d_in[9]};
  const float* b1[2]  = {(const float*)d_in[6],  (const float*)d_in[10]};
  const float* W2[2]  = {(const float*)d_in[11], (const float*)d_in[15]};
  const float* as2[2] = {(const float*)d_in[12], (const float*)d_in[16]};
  const float* ad2[2] = {(const float*)d_in[13], (const float*)d_in[17]};
  const float* b2[2]  = {(const float*)d_in[14], (const float*)d_in[18]};
  const float* w_lin  = (const float*)d_in[19];
  const float* b_lin  = (const float*)d_in[20];

  const int n  = NNODES;
  const int E  = in_sizes[1] / 2;   // 65536 edges per relation (pre self-loop)
  const int nE = E + n;

  // workspace layout (floats); total ~5.3 MB
  float* ws     = (float*)d_ws;
  float* xp     = ws; ws += (size_t)n * 512;   // per-relation projection (reused)
  float* h1     = ws; ws += (size_t)n * 512;
  float* h2     = ws; ws += (size_t)n * 128;
  float* al_s   = ws; ws += n * NH;
  float* al_d   = ws; ws += n * NH;
  float* mseg   = ws; ws += n * NH;
  float* sseg   = ws; ws += n * NH;
  float* coef   = ws; ws += (size_t)nE * NH;
  float* s_node = ws; ws += n;

  // ---------------- layer 1: 256 -> H*256 = 512, summed over 2 relations --------
  fill_kernel<<<nblk(n * 512, 256), 256, 0, stream>>>(h1, 0.f, n * 512);
  for (int r = 0; r < 2; ++r) {
    const int tiles = (n / 16) * (512 / 16);
    gemm_wmma_f32<<<nblk(tiles, 8), 256, 0, stream>>>(x, W1[r], xp, n, 256, 512);
    compute_al<<<nblk(n * NH, 256), 256, 0, stream>>>(xp, as1[r], ad1[r], al_s, al_d, n, 256);
    fill_kernel<<<nblk(n * NH, 256), 256, 0, stream>>>(mseg, -INFINITY, n * NH);
    fill_kernel<<<nblk(n * NH, 256), 256, 0, stream>>>(sseg, 0.f, n * NH);
    edge_max <<<nblk(nE * NH, 256), 256, 0, stream>>>(ei[r], E, n, al_s, al_d, mseg);
    edge_sum <<<nblk(nE * NH, 256), 256, 0, stream>>>(ei[r], E, n, al_s, al_d, mseg, coef, sseg);
    coef_norm<<<nblk(nE * NH, 256), 256, 0, stream>>>(ei[r], E, n, coef, sseg);
    edge_agg <<<nblk(nE * 512, 256), 256, 0, stream>>>(ei[r], E, n, 256, xp, coef, h1);
  }
  bias_act<<<nblk(n * 512, 256), 256, 0, stream>>>(h1, b1[0], b1[1], n * 512, 512, 1);

  // ---------------- layer 2: 512 -> H*64 = 128, summed over 2 relations ---------
  fill_kernel<<<nblk(n * 128, 256), 256, 0, stream>>>(h2, 0.f, n * 128);
  for (int r = 0; r < 2; ++r) {
    const int tiles = (n / 16) * (128 / 16);
    gemm_wmma_f32<<<nblk(tiles, 8), 256, 0, stream>>>(h1, W2[r], xp, n, 512, 128);
    compute_al<<<nblk(n * NH, 256), 256, 0, stream>>>(xp, as2[r], ad2[r], al_s, al_d, n, 64);
    fill_kernel<<<nblk(n * NH, 256), 256, 0, stream>>>(mseg, -INFINITY, n * NH);
    fill_kernel<<<nblk(n * NH, 256), 256, 0, stream>>>(sseg, 0.f, n * NH);
    edge_max <<<nblk(nE * NH, 256), 256, 0, stream>>>(ei[r], E, n, al_s, al_d, mseg);
    edge_sum <<<nblk(nE * NH, 256), 256, 0, stream>>>(ei[r], E, n, al_s, al_d, mseg, coef, sseg);
    coef_norm<<<nblk(nE * NH, 256), 256, 0, stream>>>(ei[r], E, n, coef, sseg);
    edge_agg <<<nblk(nE * 128, 256), 256, 0, stream>>>(ei[r], E, n, 64, xp, coef, h2);
  }
  bias_act<<<nblk(n * 128, 256), 256, 0, stream>>>(h2, b2[0], b2[1], n * 128, 128, 0);

  // ---------------- factorized pairwise output: out[i,j] = s[i]+s[j]+b ----------
  node_dot<<<nblk(n, 8), 256, 0, stream>>>(h2, w_lin, s_node, n, 128);
  pair_out<<<nblk(n * (n / 4), 256), 256, 0, stream>>>(s_node, b_lin, (float*)d_out, n);
}